// ClusteringModel_9637906612646
// MI455X (gfx1250) — compile-verified
//
#include <hip/hip_runtime.h>
#include <math.h>

// MI455X / gfx1250, wave32. Fused clustering-model kernel.
// wdist = p[b] + q[c] - 2*G[b,c],  G = (a.*x) @ w_dist^T  via V_WMMA_F32_16X16X4_F32.

typedef float v2f __attribute__((ext_vector_type(2)));
typedef float v8f __attribute__((ext_vector_type(8)));

#define B_ 1024
#define C_ 512
#define D_ 256
#define BM 16          // rows of B per workgroup
#define XS_LD 260      // padded LDS row stride (floats) for scaled-input tile
#define CL_LD 516      // padded LDS row stride (floats) for logit tile
#define NEG_BIG (-1.0e30f)

__device__ __forceinline__ float wave_max(float v) {
#pragma unroll
  for (int off = 16; off > 0; off >>= 1) v = fmaxf(v, __shfl_xor(v, off, 32));
  return v;
}
__device__ __forceinline__ float wave_sum(float v) {
#pragma unroll
  for (int off = 16; off > 0; off >>= 1) v += __shfl_xor(v, off, 32);
  return v;
}

// ---- prep: a = attn/sum(attn) ; q[c] = sum_d a_d * w[c,d]^2 ----
__global__ void __launch_bounds__(256) prep_kernel(const float* __restrict__ attn,
                                                   const float* __restrict__ w_dist,
                                                   float* __restrict__ ws_a,
                                                   float* __restrict__ ws_q) {
  __shared__ float aS[D_];
  __shared__ float red[8];
  const int tid = threadIdx.x;            // 256 threads
  const float v = attn[tid];
  const float s = wave_sum(v);
  if ((tid & 31) == 0) red[tid >> 5] = s;
  __syncthreads();
  float tot = 0.0f;
#pragma unroll
  for (int i = 0; i < 8; ++i) tot += red[i];
  const float a = v / tot;
  aS[tid] = a;
  ws_a[tid] = a;
  __syncthreads();
#pragma unroll 1
  for (int c = tid; c < C_; c += 256) {
    const float* wr = w_dist + (size_t)c * D_;
    float q = 0.0f;
    for (int d = 0; d < D_; ++d) {
      const float w = wr[d];
      q = fmaf(aS[d] * w, w, q);
    }
    ws_q[c] = q;
  }
}

// ---- main fused kernel: one workgroup = 16 rows of B, all 512 clusters ----
__global__ void __launch_bounds__(256) cluster_kernel(const float* __restrict__ inp,
                                                      const float* __restrict__ w_dist,
                                                      const float* __restrict__ w_assoc,
                                                      const int* __restrict__ mask,
                                                      const float* __restrict__ ws_a,
                                                      const float* __restrict__ ws_q,
                                                      float* __restrict__ out) {
  __shared__ float xs[BM * XS_LD];     // a_d * x[b,d] tile (16 x 256, padded)
  __shared__ float pS[BM];             // p[b] = sum_d a_d x^2
  __shared__ float compL[BM * CL_LD];  // masked comp logits (16 x 512, padded)

  const int tid = threadIdx.x;
  const int lane = tid & 31;
  const int wv = tid >> 5;             // wave id 0..7
  const int b0 = blockIdx.x * BM;

  if (tid < BM) pS[tid] = 0.0f;
  __syncthreads();

  // Stage 0: load & scale the 16x256 input tile, accumulate p[b].
  {
    const int m = tid >> 4;            // 0..15 row
    const int j = tid & 15;            // 0..15 column phase
    const float* xrow = inp + (size_t)(b0 + m) * D_;
    float psum = 0.0f;
#pragma unroll 4
    for (int i = 0; i < 16; ++i) {
      const int d = j + (i << 4);
      const float ad = ws_a[d];
      const float x = xrow[d];
      const float sx = ad * x;
      xs[m * XS_LD + d] = sx;
      psum = fmaf(sx, x, psum);        // a_d * x^2
    }
    atomicAdd(&pS[m], psum);           // ds_add_f32
  }
  __syncthreads();

  // Stage 1: GEMM  G[m][c] = sum_d xs[m][d] * w_dist[c][d]
  // V_WMMA_F32_16X16X4_F32 layouts (wave32):
  //   A 16x4 : lane L holds M=L&15, K = 2*(L>>4) + {0,1} in v[0],v[1]
  //   B 4x16 : lane L holds N=L&15, K = 2*(L>>4) + {0,1} in v[0],v[1]
  //   C/D    : reg r, lane L -> M = r + 8*(L>>4), N = L&15
  const int half = lane >> 4;          // 0/1
  const int l16 = lane & 15;
  const int cw = wv * 64;              // this wave's cluster-column base

  const float* abase = &xs[l16 * XS_LD + (half << 1)];
  const float* wr0 = w_dist + (size_t)(cw + 0 * 16 + l16) * D_ + (half << 1);
  const float* wr1 = w_dist + (size_t)(cw + 1 * 16 + l16) * D_ + (half << 1);
  const float* wr2 = w_dist + (size_t)(cw + 2 * 16 + l16) * D_ + (half << 1);
  const float* wr3 = w_dist + (size_t)(cw + 3 * 16 + l16) * D_ + (half << 1);

  v8f acc0 = {}, acc1 = {}, acc2 = {}, acc3 = {};
#pragma unroll 2
  for (int kt = 0; kt < D_ / 4; ++kt) {
    const v2f af = *(const v2f*)(abase + kt * 4);
    const v2f bf0 = *(const v2f*)(wr0 + kt * 4);
    const v2f bf1 = *(const v2f*)(wr1 + kt * 4);
    const v2f bf2 = *(const v2f*)(wr2 + kt * 4);
    const v2f bf3 = *(const v2f*)(wr3 + kt * 4);
    acc0 = __builtin_amdgcn_wmma_f32_16x16x4_f32(false, af, false, bf0, (short)0, acc0, false, false);
    acc1 = __builtin_amdgcn_wmma_f32_16x16x4_f32(false, af, false, bf1, (short)0, acc1, false, false);
    acc2 = __builtin_amdgcn_wmma_f32_16x16x4_f32(false, af, false, bf2, (short)0, acc2, false, false);
    acc3 = __builtin_amdgcn_wmma_f32_16x16x4_f32(false, af, false, bf3, (short)0, acc3, false, false);
  }

  // Epilogue: wdist -> masked competition logits into LDS (reference NEG_INF semantics).
  {
    float pr[8];
#pragma unroll
    for (int r = 0; r < 8; ++r) pr[r] = pS[r + (half << 3)];

    auto epi = [&](const v8f& acc, int t) {
      const int c = cw + t * 16 + l16;
      const float qc = ws_q[c];
      const bool mk = (mask[c] != 0);
#pragma unroll
      for (int r = 0; r < 8; ++r) {
        const float wd = pr[r] + qc - 2.0f * acc[r];
        const float tt = sqrtf(fmaxf(wd, 0.0f));             // wdist^(q/r), q/r = 0.5
        const float lg = mk ? (-3.0f * tt) : NEG_BIG;        // beta*log(H)/temp = -3*sqrt(wdist)
        compL[(r + (half << 3)) * CL_LD + c] = lg;
      }
    };
    epi(acc0, 0); epi(acc1, 1); epi(acc2, 2); epi(acc3, 3);
  }
  __syncthreads();

  // Stage 2: per-row softmax chains + readout. Wave wv owns rows 2wv, 2wv+1.
#pragma unroll
  for (int rr = 0; rr < 2; ++rr) {
    const int m = wv * 2 + rr;
    float* row = &compL[m * CL_LD];

    // ClusterCompetition softmax over logits (masked entries carry -1e30, like reference)
    float mx = NEG_BIG;
    for (int c = lane; c < C_; c += 32) mx = fmaxf(mx, row[c]);
    mx = wave_max(mx);
    float sm = 0.0f;
    for (int c = lane; c < C_; c += 32) sm += __expf(row[c] - mx);
    sm = wave_sum(sm);
    const float inv = 1.0f / sm;

    // competed = p_comp * H * mask ; start SoftWTA max. Overwrite row with competed
    // (masked slots -> -1 sentinel; competed >= 0 always).
    float mx2 = NEG_BIG;
    for (int c = lane; c < C_; c += 32) {
      const float lg = row[c];
      const bool mk = (lg > -1.0e29f);
      const float pcomp = __expf(lg - mx) * inv;
      const float H = mk ? __expf(lg * (1.0f / 3.0f)) : 0.0f;  // H = exp(-t) = exp(lg/3)
      const float comp = pcomp * H;
      const float wl = mk ? comp * 10.0f : NEG_BIG;            // competed / TEMP_WTA
      mx2 = fmaxf(mx2, wl);
      row[c] = mk ? comp : -1.0f;
    }
    mx2 = wave_max(mx2);

    float s2 = 0.0f;
    for (int c = lane; c < C_; c += 32) {
      const float comp = row[c];
      const float wl = (comp >= 0.0f) ? comp * 10.0f : NEG_BIG;
      s2 += __expf(wl - mx2);
    }
    s2 = wave_sum(s2);
    const float inv2 = 1.0f / s2;

    // softwta = p_wta * competed ; y = PHI * softwta @ w_assoc
    float y0 = 0.0f, y1 = 0.0f;
    for (int c = lane; c < C_; c += 32) {
      const float comp = row[c];
      const bool mk = (comp >= 0.0f);
      const float cc = mk ? comp : 0.0f;
      const float wl = mk ? comp * 10.0f : NEG_BIG;
      const float sw = __expf(wl - mx2) * inv2 * cc;
      y0 = fmaf(sw, w_assoc[c * 2 + 0], y0);
      y1 = fmaf(sw, w_assoc[c * 2 + 1], y1);
    }
    y0 = wave_sum(y0);
    y1 = wave_sum(y1);
    if (lane == 0) {
      out[(size_t)(b0 + m) * 2 + 0] = 1.5f * y0;
      out[(size_t)(b0 + m) * 2 + 1] = 1.5f * y1;
    }
  }
}

extern "C" void kernel_launch(void* const* d_in, const int* in_sizes, int n_in,
                              void* d_out, int out_size, void* d_ws, size_t ws_size,
                              hipStream_t stream) {
  const float* inp = (const float*)d_in[0];      // [B, D]
  const float* w_dist = (const float*)d_in[1];   // [C, D]
  const float* attn = (const float*)d_in[2];     // [D]
  const float* w_assoc = (const float*)d_in[3];  // [C, 2]
  const int* mask = (const int*)d_in[4];         // [C]
  float* out = (float*)d_out;                    // [B, 2]

  float* ws_a = (float*)d_ws;                    // 256 floats
  float* ws_q = ws_a + D_;                       // 512 floats

  prep_kernel<<<1, 256, 0, stream>>>(attn, w_dist, ws_a, ws_q);
  cluster_kernel<<<B_ / BM, 256, 0, stream>>>(inp, w_dist, w_assoc, mask, ws_a, ws_q, out);
}